// PointNetSegCls_60687887892803
// MI455X (gfx1250) — compile-verified
//
#include <hip/hip_runtime.h>
#include <hip/hip_bf16.h>

typedef float v2f __attribute__((ext_vector_type(2)));
typedef float v8f __attribute__((ext_vector_type(8)));

#define BN 16
#define NPTS 8192
#define TOPK 4096

// ---------------------------------------------------------------------------
// c1: x1[b,o,n] = relu(s[o]*(W[o,:]·p[b,n,:]) + b[o]),  W is 64x3
// ---------------------------------------------------------------------------
__global__ __launch_bounds__(256) void c1_kernel(
    const float* __restrict__ pts, const float* __restrict__ w,
    const float* __restrict__ sc, const float* __restrict__ bc,
    float* __restrict__ x1, int N)
{
    int n = blockIdx.x * 256 + threadIdx.x;
    int b = blockIdx.y;
    const float* p = pts + ((size_t)b * N + n) * 3;
    float p0 = p[0], p1 = p[1], p2 = p[2];
    float* out = x1 + (size_t)b * 64 * N + n;
#pragma unroll 4
    for (int o = 0; o < 64; ++o) {
        float v = w[o * 3 + 0] * p0 + w[o * 3 + 1] * p1 + w[o * 3 + 2] * p2;
        v = fmaxf(v * sc[o] + bc[o], 0.0f);
        out[(size_t)o * N] = v;
    }
}

// ---------------------------------------------------------------------------
// Generic WMMA f32 GEMM: Y[b] = relu(scale*(W·X[b] + cvec[b]) + bias)
//   W: M x K (row stride ldw), X: K x N per batch, Y: M x N per batch
//   Block: 128 threads = 4 waves; tile 64(M) x 64(N); each wave 32x32.
//   Uses V_WMMA_F32_16X16X4_F32, K staged via LDS in chunks of 16.
// ---------------------------------------------------------------------------
__global__ __launch_bounds__(128) void gemm_relu_wmma(
    const float* __restrict__ W, int ldw,
    const float* __restrict__ X, float* __restrict__ Y,
    const float* __restrict__ scale, const float* __restrict__ bias,
    const float* __restrict__ cvec,   // nullable, [b][M] additive pre-scale
    int M, int K, int N)
{
    const int b  = blockIdx.z;
    const int n0 = blockIdx.x * 64;
    const int m0 = blockIdx.y * 64;
    const float* Xb = X + (size_t)b * K * N;
    float*       Yb = Y + (size_t)b * M * N;

    __shared__ float As[64][17];   // M x K-chunk (padded vs bank conflicts)
    __shared__ float Bs[16][65];   // K-chunk x N

    const int tid  = threadIdx.x;
    const int wave = tid >> 5;
    const int lane = tid & 31;
    const int wm   = (wave >> 1) * 32;   // wave's M offset in tile
    const int wn   = (wave & 1)  * 32;   // wave's N offset in tile
    const int l15  = lane & 15;
    const int lhi  = lane >> 4;          // 0 -> K{0,1}, 1 -> K{2,3}

    v8f acc[2][2];
#pragma unroll
    for (int i = 0; i < 2; ++i)
#pragma unroll
        for (int j = 0; j < 2; ++j)
            acc[i][j] = (v8f){0.f, 0.f, 0.f, 0.f, 0.f, 0.f, 0.f, 0.f};

    for (int k0 = 0; k0 < K; k0 += 16) {
        // stage W tile: 64x16
#pragma unroll
        for (int i = tid; i < 64 * 16; i += 128) {
            int m = i >> 4, k = i & 15;
            As[m][k] = W[(size_t)(m0 + m) * ldw + (k0 + k)];
        }
        // stage X tile: 16x64
#pragma unroll
        for (int i = tid; i < 16 * 64; i += 128) {
            int k = i >> 6, n = i & 63;
            Bs[k][n] = Xb[(size_t)(k0 + k) * N + (n0 + n)];
        }
        __syncthreads();

#pragma unroll
        for (int kk = 0; kk < 16; kk += 4) {
            v2f afrag[2], bfrag[2];
#pragma unroll
            for (int i = 0; i < 2; ++i) {
                int m = wm + i * 16 + l15;
                afrag[i].x = As[m][kk + lhi * 2 + 0];
                afrag[i].y = As[m][kk + lhi * 2 + 1];
                int n = wn + i * 16 + l15;
                bfrag[i].x = Bs[kk + lhi * 2 + 0][n];
                bfrag[i].y = Bs[kk + lhi * 2 + 1][n];
            }
#pragma unroll
            for (int i = 0; i < 2; ++i)
#pragma unroll
                for (int j = 0; j < 2; ++j)
                    acc[i][j] = __builtin_amdgcn_wmma_f32_16x16x4_f32(
                        false, afrag[i], false, bfrag[j],
                        (short)0, acc[i][j], false, false);
        }
        __syncthreads();
    }

    // Epilogue. C/D layout: VGPR r -> (M = base + lhi*8 + r, N = base + l15)
#pragma unroll
    for (int i = 0; i < 2; ++i) {
#pragma unroll
        for (int j = 0; j < 2; ++j) {
            int mbase = m0 + wm + i * 16 + lhi * 8;
            int nn    = n0 + wn + j * 16 + l15;
#pragma unroll
            for (int r = 0; r < 8; ++r) {
                int m = mbase + r;
                float v = acc[i][j][r];
                if (cvec) v += cvec[(size_t)b * M + m];
                v = fmaxf(v * scale[m] + bias[m], 0.0f);
                Yb[(size_t)m * N + nn] = v;
            }
        }
    }
}

// ---------------------------------------------------------------------------
// Global max over points: g[b,m] = max_n x4[b,m,n]
// ---------------------------------------------------------------------------
__global__ __launch_bounds__(256) void gmax_kernel(
    const float* __restrict__ x4, float* __restrict__ g, int N)
{
    int m = blockIdx.x, b = blockIdx.y;
    const float* p = x4 + ((size_t)b * 512 + m) * N;
    float mx = -3.402823466e38f;
    for (int n = threadIdx.x; n < N; n += 256) mx = fmaxf(mx, p[n]);
    __shared__ float red[256];
    red[threadIdx.x] = mx;
    __syncthreads();
    for (int off = 128; off > 0; off >>= 1) {
        if (threadIdx.x < off)
            red[threadIdx.x] = fmaxf(red[threadIdx.x], red[threadIdx.x + off]);
        __syncthreads();
    }
    if (threadIdx.x == 0) g[b * 512 + m] = red[0];
}

// ---------------------------------------------------------------------------
// cvec[b,m] = s1_w[m, 128:640] · g[b, :]   (broadcast-g contribution to s1)
// ---------------------------------------------------------------------------
__global__ __launch_bounds__(256) void sconst_kernel(
    const float* __restrict__ s1w, const float* __restrict__ g,
    float* __restrict__ cvec)
{
    int m = threadIdx.x, b = blockIdx.x;
    const float* wr = s1w + (size_t)m * 640 + 128;
    const float* gb = g + b * 512;
    float a = 0.f;
#pragma unroll 8
    for (int k = 0; k < 512; ++k) a += wr[k] * gb[k];
    cvec[b * 256 + m] = a;
}

// ---------------------------------------------------------------------------
// s3 + softmax: seg_logits -> d_out, score = P(class 1)
// ---------------------------------------------------------------------------
__global__ __launch_bounds__(256) void s3_kernel(
    const float* __restrict__ h2, const float* __restrict__ w,
    const float* __restrict__ bias, float* __restrict__ seg_out,
    float* __restrict__ score, int N)
{
    int n = blockIdx.x * 256 + threadIdx.x;
    int b = blockIdx.y;
    const float* col = h2 + (size_t)b * 128 * N + n;
    float l0 = 0.f, l1 = 0.f;
#pragma unroll 8
    for (int k = 0; k < 128; ++k) {
        float x = col[(size_t)k * N];
        l0 += w[k] * x;
        l1 += w[128 + k] * x;
    }
    l0 += bias[0]; l1 += bias[1];
    seg_out[((size_t)b * 2 + 0) * N + n] = l0;
    seg_out[((size_t)b * 2 + 1) * N + n] = l1;
    float m  = fmaxf(l0, l1);
    float e0 = __expf(l0 - m), e1 = __expf(l1 - m);
    score[(size_t)b * N + n] = e1 / (e0 + e1);
}

// ---------------------------------------------------------------------------
// Radix bisection on score bits: largest t with count(score >= t) >= k.
// Scores are in (0,1) so uint bit ordering == float ordering.
// ---------------------------------------------------------------------------
__global__ __launch_bounds__(256) void thresh_kernel(
    const float* __restrict__ score, float* __restrict__ thr, int N, int k)
{
    int b = blockIdx.x;
    const float* sc = score + (size_t)b * N;
    __shared__ int red[256];
    unsigned cur = 0u;
    for (int bit = 31; bit >= 0; --bit) {
        unsigned cand = cur | (1u << bit);
        int c = 0;
        for (int n = threadIdx.x; n < N; n += 256)
            if (__float_as_uint(sc[n]) >= cand) c++;
        red[threadIdx.x] = c;
        __syncthreads();
        for (int off = 128; off > 0; off >>= 1) {
            if (threadIdx.x < off) red[threadIdx.x] += red[threadIdx.x + off];
            __syncthreads();
        }
        int total = red[0];
        __syncthreads();
        if (total >= k) cur = cand;
    }
    if (threadIdx.x == 0) thr[b] = __uint_as_float(cur);
}

// ---------------------------------------------------------------------------
// Masked max/mean pooling of x4 + assemble cls_feat = [max | mean | g]
// ---------------------------------------------------------------------------
__global__ __launch_bounds__(256) void pool_kernel(
    const float* __restrict__ x4, const float* __restrict__ score,
    const float* __restrict__ thr, const float* __restrict__ g,
    float* __restrict__ feat, int N)
{
    int m = blockIdx.x, b = blockIdx.y;
    const float* p  = x4 + ((size_t)b * 512 + m) * N;
    const float* sc = score + (size_t)b * N;
    unsigned tu = __float_as_uint(thr[b]);
    float smax = -3.402823466e38f, amax = -3.402823466e38f, ssum = 0.f;
    int scnt = 0;
    for (int n = threadIdx.x; n < N; n += 256) {
        float v = p[n];
        amax = fmaxf(amax, v);
        if (__float_as_uint(sc[n]) >= tu) {
            smax = fmaxf(smax, v);
            ssum += v;
            scnt++;
        }
    }
    __shared__ float rsm[256], rsa[256], rss[256];
    __shared__ int   rsc[256];
    rsm[threadIdx.x] = smax; rsa[threadIdx.x] = amax;
    rss[threadIdx.x] = ssum; rsc[threadIdx.x] = scnt;
    __syncthreads();
    for (int off = 128; off > 0; off >>= 1) {
        if (threadIdx.x < off) {
            rsm[threadIdx.x] = fmaxf(rsm[threadIdx.x], rsm[threadIdx.x + off]);
            rsa[threadIdx.x] = fmaxf(rsa[threadIdx.x], rsa[threadIdx.x + off]);
            rss[threadIdx.x] += rss[threadIdx.x + off];
            rsc[threadIdx.x] += rsc[threadIdx.x + off];
        }
        __syncthreads();
    }
    if (threadIdx.x == 0) {
        int   cnt  = rsc[0];
        float mx   = (cnt > 0) ? rsm[0] : rsa[0];
        float mean = rss[0] / fmaxf((float)cnt, 1e-6f);
        feat[(size_t)b * 1536 + m]        = mx;
        feat[(size_t)b * 1536 + 512 + m]  = mean;
        feat[(size_t)b * 1536 + 1024 + m] = g[b * 512 + m];
    }
}

// ---------------------------------------------------------------------------
// Classifier head: 1536 -> 256 -> 128 -> 32 (one block per batch row)
// ---------------------------------------------------------------------------
__global__ __launch_bounds__(256) void head_kernel(
    const float* __restrict__ feat,
    const float* __restrict__ h1w, const float* __restrict__ h1s, const float* __restrict__ h1b,
    const float* __restrict__ h2w, const float* __restrict__ h2s, const float* __restrict__ h2b,
    const float* __restrict__ h3w, const float* __restrict__ h3b,
    float* __restrict__ out)
{
    int b = blockIdx.x, t = threadIdx.x;
    __shared__ float f[1536];
    __shared__ float f1[256];
    __shared__ float f2[128];
    for (int i = t; i < 1536; i += 256) f[i] = feat[(size_t)b * 1536 + i];
    __syncthreads();
    {
        const float* wr = h1w + (size_t)t * 1536;
        float a = 0.f;
        for (int k = 0; k < 1536; ++k) a += wr[k] * f[k];
        f1[t] = fmaxf(a * h1s[t] + h1b[t], 0.0f);
    }
    __syncthreads();
    if (t < 128) {
        const float* wr = h2w + (size_t)t * 256;
        float a = 0.f;
        for (int k = 0; k < 256; ++k) a += wr[k] * f1[k];
        f2[t] = fmaxf(a * h2s[t] + h2b[t], 0.0f);
    }
    __syncthreads();
    if (t < 32) {
        const float* wr = h3w + (size_t)t * 128;
        float a = 0.f;
        for (int k = 0; k < 128; ++k) a += wr[k] * f2[k];
        out[b * 32 + t] = a + h3b[t];
    }
}

// ---------------------------------------------------------------------------
extern "C" void kernel_launch(void* const* d_in, const int* in_sizes, int n_in,
                              void* d_out, int out_size, void* d_ws, size_t ws_size,
                              hipStream_t stream) {
    (void)in_sizes; (void)n_in; (void)out_size; (void)ws_size;
    const int N = NPTS, B = BN;

    const float* pts   = (const float*)d_in[0];
    const float* c1_w  = (const float*)d_in[1];
    const float* c1_s  = (const float*)d_in[2];
    const float* c1_b  = (const float*)d_in[3];
    const float* c2_w  = (const float*)d_in[4];
    const float* c2_s  = (const float*)d_in[5];
    const float* c2_b  = (const float*)d_in[6];
    const float* c3_w  = (const float*)d_in[7];
    const float* c3_s  = (const float*)d_in[8];
    const float* c3_b  = (const float*)d_in[9];
    const float* c4_w  = (const float*)d_in[10];
    const float* c4_s  = (const float*)d_in[11];
    const float* c4_b  = (const float*)d_in[12];
    const float* s1_w  = (const float*)d_in[13];
    const float* s1_s  = (const float*)d_in[14];
    const float* s1_b  = (const float*)d_in[15];
    const float* s2_w  = (const float*)d_in[16];
    const float* s2_s  = (const float*)d_in[17];
    const float* s2_b  = (const float*)d_in[18];
    const float* s3_w  = (const float*)d_in[19];
    const float* s3_bb = (const float*)d_in[20];
    const float* h1_w  = (const float*)d_in[21];
    const float* h1_s  = (const float*)d_in[22];
    const float* h1_b  = (const float*)d_in[23];
    const float* h2_w  = (const float*)d_in[24];
    const float* h2_s  = (const float*)d_in[25];
    const float* h2_b  = (const float*)d_in[26];
    const float* h3_w  = (const float*)d_in[27];
    const float* h3_bb = (const float*)d_in[28];

    float* ws = (float*)d_ws;
    // workspace layout (floats); x3 region reused for s1-out, x2 region for s2-out
    float* x1    = ws;                       //  8,388,608 : [B][ 64][N]
    float* x2    = ws + 8388608;             // 16,777,216 : [B][128][N]
    float* x3    = ws + 25165824;            // 33,554,432 : [B][256][N]
    float* x4    = ws + 58720256;            // 67,108,864 : [B][512][N]
    float* gbuf  = ws + 125829120;           //      8,192 : [B][512]
    float* cvec  = gbuf + 8192;              //      4,096 : [B][256]
    float* score = cvec + 4096;              //    131,072 : [B][N]
    float* thr   = score + 131072;           //         16
    float* feat  = thr + 16;                 //     24,576 : [B][1536]
    float* hbuf  = x3;                       // s1 output  : [B][256][N]
    float* h2buf = x2;                       // s2 output  : [B][128][N]

    float* cls_out = (float*)d_out;          // [B][32]
    float* seg_out = cls_out + B * 32;       // [B][2][N]

    c1_kernel<<<dim3(N / 256, B), 256, 0, stream>>>(pts, c1_w, c1_s, c1_b, x1, N);

    gemm_relu_wmma<<<dim3(N / 64, 2, B), 128, 0, stream>>>(
        c2_w, 64, x1, x2, c2_s, c2_b, nullptr, 128, 64, N);
    gemm_relu_wmma<<<dim3(N / 64, 4, B), 128, 0, stream>>>(
        c3_w, 128, x2, x3, c3_s, c3_b, nullptr, 256, 128, N);
    gemm_relu_wmma<<<dim3(N / 64, 8, B), 128, 0, stream>>>(
        c4_w, 256, x3, x4, c4_s, c4_b, nullptr, 512, 256, N);

    gmax_kernel<<<dim3(512, B), 256, 0, stream>>>(x4, gbuf, N);
    sconst_kernel<<<B, 256, 0, stream>>>(s1_w, gbuf, cvec);

    gemm_relu_wmma<<<dim3(N / 64, 4, B), 128, 0, stream>>>(
        s1_w, 640, x2, hbuf, s1_s, s1_b, cvec, 256, 128, N);
    gemm_relu_wmma<<<dim3(N / 64, 2, B), 128, 0, stream>>>(
        s2_w, 256, hbuf, h2buf, s2_s, s2_b, nullptr, 128, 256, N);

    s3_kernel<<<dim3(N / 256, B), 256, 0, stream>>>(h2buf, s3_w, s3_bb, seg_out, score, N);
    thresh_kernel<<<B, 256, 0, stream>>>(score, thr, N, TOPK);
    pool_kernel<<<dim3(512, B), 256, 0, stream>>>(x4, score, thr, gbuf, feat, N);
    head_kernel<<<B, 256, 0, stream>>>(feat, h1_w, h1_s, h1_b,
                                       h2_w, h2_s, h2_b, h3_w, h3_bb, cls_out);
}